// NNConvModel_28217935134974
// MI455X (gfx1250) — compile-verified
//
#include <hip/hip_runtime.h>
#include <hip/hip_bf16.h>

// NNConv GNN reference: the returned tensor is ONLY `e @ pred_W + pred_b`
// ([E,19]x[19,2] + [2]) -- the 3-step message-passing loop never feeds the
// output and is dead code. We compute just the edge predictor, streaming
// e (15.2 MB) once: bandwidth-bound, ~0.72us at 23.3 TB/s on MI455X.
//
// Matrix pipe: V_WMMA_F32_16X16X4_F32 (f32 in/out, exact reference precision).
// Per wave: one 16-edge tile, K=19 padded to 20 -> 5 chained WMMAs,
// N=2 padded to 16 (B columns 2..15 zero). Hot loop has NO guarded loads:
//  - the K=19 pad lane reads the next row's element 0 (finite) and its B
//    entry is zero, so it contributes exactly 0 to the accumulator;
//  - tiles are split into "safe" tiles (every touched byte in-bounds:
//    t*16+17 <= E) processed guard-free, plus a <=16-row scalar tail.

typedef __attribute__((ext_vector_type(2))) float v2f;
typedef __attribute__((ext_vector_type(8))) float v8f;

__global__ __launch_bounds__(256) void edge_pred_wmma_f32(
    const float* __restrict__ efeat,   // [E, 19] row-major
    const float* __restrict__ predW,   // [19, 2] row-major
    const float* __restrict__ predB,   // [2]
    float* __restrict__ out,           // [E, 2] row-major
    int E,
    int safeTiles)
{
    const int lane   = threadIdx.x & 31;
    const int wave   = (int)((blockIdx.x * blockDim.x + threadIdx.x) >> 5);
    const int nwaves = (int)((gridDim.x * blockDim.x) >> 5);

    // WMMA f32 16x16x4 operand mapping (wave32):
    //   A (16x4):  lane l -> row m = l&15 ; VGPR0 = A[m][kb], VGPR1 = A[m][kb+1],
    //              kb = 2*(l>>4)
    //   B (4x16):  lane l -> col n = l&15 ; VGPR0 = B[kb][n], VGPR1 = B[kb+1][n]
    //   C/D (16x16): VGPR v of lane l -> D[v + 8*(l>>4)][l&15]
    const int mn = lane & 15;          // A-row index == B/D-column index
    const int kb = (lane >> 4) << 1;   // K sub-offset within a 4-wide K step

    // Loop-invariant B fragments: K padded 19->20, N padded 2->16 with zeros.
    // bfrag[4].y == 0 for kb==2 lanes (k=19) -> pad A element contributes 0.
    v2f bfrag[5];
#pragma unroll
    for (int s = 0; s < 5; ++s) {
        const int k0 = s * 4 + kb;
        bfrag[s].x = (mn < 2 && k0     < 19) ? predW[(size_t)k0 * 2 + mn]       : 0.0f;
        bfrag[s].y = (mn < 2 && k0 + 1 < 19) ? predW[(size_t)(k0 + 1) * 2 + mn] : 0.0f;
    }
    const float biasn = (mn < 2) ? predB[mn] : 0.0f;

    // ---- Main loop: guard-free tiles, fully pipelined unconditional loads ----
    for (int t = wave; t < safeTiles; t += nwaves) {
        const float* base = efeat + (size_t)(t * 16 + mn) * 19 + kb;

        v8f acc = {};
#pragma unroll
        for (int s = 0; s < 5; ++s) {
            v2f a;
            a.x = base[4 * s];         // k = 4s + kb
            a.y = base[4 * s + 1];     // k = 4s + kb + 1 (k=19 is harmless pad)
            // 8 args: (neg_a, A, neg_b, B, c_mod, C, reuse_a, reuse_b)
            acc = __builtin_amdgcn_wmma_f32_16x16x4_f32(
                false, a, false, bfrag[s], (short)0, acc, false, false);
        }

        // Columns 0/1 live in lanes 0,1,16,17; all 16 rows valid in safe tiles.
        if (mn < 2) {
            const int rowBase = t * 16 + ((lane >> 4) << 3);
#pragma unroll
            for (int v = 0; v < 8; ++v) {
                out[(size_t)(rowBase + v) * 2 + mn] = acc[v] + biasn;
            }
        }
    }

    // ---- Tail: <=16 rows, plain per-lane dot products (no WMMA, EXEC-safe) ----
    const int tailStart = safeTiles * 16;
    if (wave == 0 && tailStart < E) {
        const int r = tailStart + lane;
        if (r < E) {
            const float* erow = efeat + (size_t)r * 19;
            float acc0 = 0.0f, acc1 = 0.0f;
#pragma unroll
            for (int k = 0; k < 19; ++k) {
                const float v = erow[k];
                acc0 += v * predW[k * 2 + 0];
                acc1 += v * predW[k * 2 + 1];
            }
            out[(size_t)r * 2 + 0] = acc0 + predB[0];
            out[(size_t)r * 2 + 1] = acc1 + predB[1];
        }
    }
}

extern "C" void kernel_launch(void* const* d_in, const int* in_sizes, int n_in,
                              void* d_out, int out_size, void* d_ws, size_t ws_size,
                              hipStream_t stream) {
    (void)d_ws; (void)ws_size;

    const int E = out_size / 2;  // output is [E, 2] float32

    // Positional defaults per setup_inputs() dict order (pred_W / pred_b are
    // the final two param leaves), with a size-based scan for robustness:
    // pred_W has a unique flat size of 19*2 = 38, pred_b of 2, e of E*19.
    const float* efeat = (const float*)d_in[2];
    const float* predW = (const float*)d_in[n_in - 2];
    const float* predB = (const float*)d_in[n_in - 1];
    for (int i = 0; i < n_in; ++i) {
        if (in_sizes[i] == E * 19) efeat = (const float*)d_in[i];
        if (in_sizes[i] == 38)     predW = (const float*)d_in[i];
        if (in_sizes[i] == 2)      predB = (const float*)d_in[i];
    }

    // Tiles where every touched byte (incl. the k=19 pad element of the last
    // row) is in-bounds: need t*16 + 17 <= E.
    const int safeTiles = (E >= 17) ? ((E - 17) / 16 + 1) : 0;

    const int threads       = 256;            // 8 wave32 waves per block
    const int wavesPerBlock = threads / 32;
    int blocks = (safeTiles + wavesPerBlock - 1) / wavesPerBlock;
    if (blocks < 1)    blocks = 1;
    if (blocks > 2048) blocks = 2048;         // grid-stride amortizes B setup

    edge_pred_wmma_f32<<<blocks, threads, 0, stream>>>(
        efeat, predW, predB, (float*)d_out, E, safeTiles);
}